// CompressModel_9740985827581
// MI455X (gfx1250) — compile-verified
//
#include <hip/hip_runtime.h>

typedef __attribute__((ext_vector_type(2))) float v2f;
typedef __attribute__((ext_vector_type(4))) float v4f;
typedef __attribute__((ext_vector_type(8))) float v8f;

#define DIM 1024
#define LSTRIDE 33   // padded LDS row stride (floats) to avoid bank conflicts

__device__ __forceinline__ float hsign(int a, int b) {
    // Sylvester Hadamard entry: (-1)^popcount(a & b)
    return (__popc(a & b) & 1) ? -1.0f : 1.0f;
}

__global__ __launch_bounds__(256)
void fht_golay_wmma_kernel(const float* __restrict__ x,
                           const float* __restrict__ golay,
                           const float* __restrict__ alpha,
                           float* __restrict__ out,
                           int nrows) {
    __shared__ float lds[8 * 32 * LSTRIDE];

    const int tid  = threadIdx.x;
    const int wave = tid >> 5;
    const int lane = tid & 31;
    const int row  = blockIdx.x * 8 + wave;
    if (row >= nrows) return;            // whole-wave uniform; active waves keep EXEC all-1s

    float* X = &lds[wave * 32 * LSTRIDE];
    const float scale = 1.0f / (alpha[0] + 1e-5f);
    const float* __restrict__ xr = x + (size_t)row * DIM;
    float* __restrict__ orow = out + (size_t)row * DIM;

    const int h = lane >> 4;             // lane half (0/1)
    const int m = lane & 15;             // M (A) / N (B) index within fragment

    // ---- Stage 1: coalesced load, multiply by golay, stash 32x32 tile in LDS ----
    #pragma unroll
    for (int it = 0; it < 8; ++it) {
        int pos = it * 128 + lane * 4;   // 512 contiguous bytes per wave per iter
        v4f xv = *(const v4f*)(xr + pos);
        v4f gv = *(const v4f*)(golay + pos);
        int i = pos >> 5, j = pos & 31;
        float* dst = X + i * LSTRIDE + j;
        dst[0] = xv.x * gv.x;
        dst[1] = xv.y * gv.y;
        dst[2] = xv.z * gv.z;
        dst[3] = xv.w * gv.w;
    }
    __builtin_amdgcn_wave_barrier();

    // ---- Stage 2: T = H32 * X  via  T0q = H16*(Xt+Xb), T1q = H16*(Xt-Xb) ----
    v8f T00, T01, T10, T11;
    #pragma unroll
    for (int q = 0; q < 2; ++q) {
        v8f t0 = {}; v8f t1 = {};
        #pragma unroll
        for (int k = 0; k < 4; ++k) {
            int kk0 = 4 * k + 2 * h;     // K index for fragment VGPR 0
            int kk1 = kk0 + 1;           // K index for fragment VGPR 1
            // A fragment = H16[:, kk] (computed, never loaded)
            v2f A; A.x = hsign(m, kk0); A.y = hsign(m, kk1);
            // B fragments = (Xtop +/- Xbot)[kk][q*16 + n]
            int c = q * 16 + m;
            float t0r0 = X[kk0 * LSTRIDE + c];
            float b0r0 = X[(kk0 + 16) * LSTRIDE + c];
            float t0r1 = X[kk1 * LSTRIDE + c];
            float b0r1 = X[(kk1 + 16) * LSTRIDE + c];
            v2f Bs; Bs.x = t0r0 + b0r0; Bs.y = t0r1 + b0r1;
            v2f Bd; Bd.x = t0r0 - b0r0; Bd.y = t0r1 - b0r1;
            t0 = __builtin_amdgcn_wmma_f32_16x16x4_f32(false, A, false, Bs, (short)0, t0, false, false);
            t1 = __builtin_amdgcn_wmma_f32_16x16x4_f32(false, A, false, Bd, (short)0, t1, false, false);
        }
        if (q == 0) { T00 = t0; T10 = t1; } else { T01 = t0; T11 = t1; }
    }

    // ---- Stage 3: spill T (C/D layout -> row-major 32x32 in LDS) ----
    __builtin_amdgcn_wave_barrier();
    #pragma unroll
    for (int r = 0; r < 8; ++r) {
        int i0 = r + 8 * h;
        X[(i0)      * LSTRIDE + m]      = T00[r];
        X[(i0)      * LSTRIDE + 16 + m] = T01[r];
        X[(i0 + 16) * LSTRIDE + m]      = T10[r];
        X[(i0 + 16) * LSTRIDE + 16 + m] = T11[r];
    }
    __builtin_amdgcn_wave_barrier();

    // ---- Stage 4: Y = T * H32  via  Yp0 = (Tp0+Tp1)*H16, Yp1 = (Tp0-Tp1)*H16 ----
    v8f Y00, Y01, Y10, Y11;
    #pragma unroll
    for (int p = 0; p < 2; ++p) {
        v8f y0 = {}; v8f y1 = {};
        #pragma unroll
        for (int k = 0; k < 4; ++k) {
            int kk0 = 4 * k + 2 * h;
            int kk1 = kk0 + 1;
            const float* trow = X + (p * 16 + m) * LSTRIDE;
            float a0l = trow[kk0],      a0r = trow[16 + kk0];
            float a1l = trow[kk1],      a1r = trow[16 + kk1];
            v2f As; As.x = a0l + a0r; As.y = a1l + a1r;
            v2f Ad; Ad.x = a0l - a0r; Ad.y = a1l - a1r;
            // B fragment = H16[kk][:] (computed)
            v2f B; B.x = hsign(kk0, m); B.y = hsign(kk1, m);
            y0 = __builtin_amdgcn_wmma_f32_16x16x4_f32(false, As, false, B, (short)0, y0, false, false);
            y1 = __builtin_amdgcn_wmma_f32_16x16x4_f32(false, Ad, false, B, (short)0, y1, false, false);
        }
        if (p == 0) { Y00 = y0; Y01 = y1; } else { Y10 = y0; Y11 = y1; }
    }

    // ---- Stage 5: spill Y to LDS, then coalesced scaled store ----
    __builtin_amdgcn_wave_barrier();
    #pragma unroll
    for (int r = 0; r < 8; ++r) {
        int i0 = r + 8 * h;
        X[(i0)      * LSTRIDE + m]      = Y00[r];
        X[(i0)      * LSTRIDE + 16 + m] = Y01[r];
        X[(i0 + 16) * LSTRIDE + m]      = Y10[r];
        X[(i0 + 16) * LSTRIDE + 16 + m] = Y11[r];
    }
    __builtin_amdgcn_wave_barrier();

    #pragma unroll
    for (int it = 0; it < 8; ++it) {
        int pos = it * 128 + lane * 4;
        int i = pos >> 5, j = pos & 31;
        const float* src = X + i * LSTRIDE + j;
        v4f o;
        o.x = src[0] * scale;
        o.y = src[1] * scale;
        o.z = src[2] * scale;
        o.w = src[3] * scale;
        *(v4f*)(orow + pos) = o;
    }
}

extern "C" void kernel_launch(void* const* d_in, const int* in_sizes, int n_in,
                              void* d_out, int out_size, void* d_ws, size_t ws_size,
                              hipStream_t stream) {
    const float* x     = (const float*)d_in[0];
    const float* golay = (const float*)d_in[1];
    const float* alpha = (const float*)d_in[2];
    float* out = (float*)d_out;

    int nrows = in_sizes[0] / DIM;           // 4*8192 = 32768
    int blocks = (nrows + 7) / 8;            // 8 waves (rows) per 256-thread block
    fht_golay_wmma_kernel<<<blocks, 256, 0, stream>>>(x, golay, alpha, out, nrows);
}